// LSTMMeasurementPredictor_75625784148625
// MI455X (gfx1250) — compile-verified
//
#include <hip/hip_runtime.h>

// Problem constants (match reference)
#define B_TOTAL 131072
#define Hh 128        // hidden
#define Tt 16         // time steps
#define DIN 17        // 1 + 8*NQ
#define Pp 32         // proj size
#define TILE 64       // batch rows per workgroup
#define NTHREADS 128  // 4 waves (wave32)

typedef float v2f __attribute__((ext_vector_type(2)));
typedef float v8f __attribute__((ext_vector_type(8)));

__device__ __forceinline__ float fast_rcp(float x) { return __builtin_amdgcn_rcpf(x); }
__device__ __forceinline__ float sigf(float x) { return fast_rcp(1.0f + __expf(-x)); }
__device__ __forceinline__ float tanha(float x) {
  return 1.0f - 2.0f * fast_rcp(__expf(2.0f * x) + 1.0f);
}

// ---- V_WMMA_F32_16X16X4_F32 fragment helpers (wave32) ----
// A (16x4): lanes 0-15 -> M=lane, K={k0,k0+1}; lanes 16-31 -> K={k0+2,k0+3}
__device__ __forceinline__ v2f frag_a(const float* __restrict__ A, int S, int rb, int k0,
                                      int lane) {
  int m = lane & 15;
  int kh = (lane >> 4) << 1;
  return *(const v2f*)(A + (rb + m) * S + k0 + kh);  // S even, kh even -> 8B aligned
}
// B (4x16) from row-major global W (Ncols, ld), ld even: W^T[k][n] = W[n][k]
__device__ __forceinline__ v2f frag_b_gl(const float* __restrict__ W, int ld, int nb, int k0,
                                         int lane) {
  int n = lane & 15;
  int kk = k0 + ((lane >> 4) << 1);
  return *(const v2f*)(W + (size_t)(nb + n) * (size_t)ld + kk);
}
// B (4x16) from LDS pair-swizzled W^T: layout [K/2][520][2] (pads banks apart)
__device__ __forceinline__ v2f frag_b_lds(const float* __restrict__ WT, int nb, int k0,
                                          int lane) {
  int n = lane & 15;
  int p = (k0 >> 1) + (lane >> 4);
  return *(const v2f*)(WT + ((p * 520) + nb + n) * 2);
}

// One LSTM gate pass: G = A1@W1^T + A2@W2^T + bsum (+ rank-1 K=16 term), gates i,f,g,o
// c := sig(f)*c + sig(i)*tanh(g);  Out := sig(o)*tanh(c).
// Wave w owns batch rows 16w..16w+15 (M-tile). Never writes A1/A2 buffers.
template <int KT1, int S1, int KT2, int S2, int SO, bool W1LDS, bool K16>
__device__ __forceinline__ void lstm_step(
    const float* __restrict__ A1, const float* __restrict__ A2,
    const float* __restrict__ W1, int ld1, const float* __restrict__ W2, int ld2,
    const float* __restrict__ bsum /*LDS[512]*/, const float* __restrict__ w16 /*LDS[512]*/,
    float* __restrict__ Cst /*[TILE][128]*/, float* __restrict__ Out /*stride SO*/) {
  const int lane = threadIdx.x & 31;
  const int wave = threadIdx.x >> 5;
  const int rb = wave * 16;
  const int m16 = lane & 15;
  const int rh = (lane >> 4) << 3;

  // cache A1 fragments (small: KT1 <= 8 -> <=16 VGPRs)
  v2f a1c[KT1];
#pragma unroll
  for (int kt = 0; kt < KT1; ++kt) a1c[kt] = frag_a(A1, S1, rb, 4 * kt, lane);

  float x16[8];
  if (K16) {
#pragma unroll
    for (int r = 0; r < 8; ++r) x16[r] = A1[(rb + rh + r) * S1 + 16];
  }

#pragma unroll 1
  for (int j = 0; j < 8; ++j) {
    v8f acc[4];
#pragma unroll
    for (int g = 0; g < 4; ++g)
#pragma unroll
      for (int e = 0; e < 8; ++e) acc[g][e] = 0.0f;

#pragma unroll
    for (int kt = 0; kt < KT1; ++kt) {
#pragma unroll
      for (int g = 0; g < 4; ++g) {
        v2f b;
        if constexpr (W1LDS) b = frag_b_lds(W1, g * Hh + j * 16, 4 * kt, lane);
        else                 b = frag_b_gl(W1, ld1, g * Hh + j * 16, 4 * kt, lane);
        acc[g] = __builtin_amdgcn_wmma_f32_16x16x4_f32(false, a1c[kt], false, b, (short)0,
                                                       acc[g], false, false);
      }
    }
#pragma unroll
    for (int kt = 0; kt < KT2; ++kt) {
      v2f a2 = frag_a(A2, S2, rb, 4 * kt, lane);
#pragma unroll
      for (int g = 0; g < 4; ++g) {
        v2f b = frag_b_gl(W2, ld2, g * Hh + j * 16, 4 * kt, lane);
        acc[g] = __builtin_amdgcn_wmma_f32_16x16x4_f32(false, a2, false, b, (short)0,
                                                       acc[g], false, false);
      }
    }

    const int n = j * 16 + m16;
    float bb0 = bsum[0 * Hh + n], bb1 = bsum[1 * Hh + n];
    float bb2 = bsum[2 * Hh + n], bb3 = bsum[3 * Hh + n];
    float wk0 = 0.f, wk1 = 0.f, wk2 = 0.f, wk3 = 0.f;
    if (K16) {
      wk0 = w16[0 * Hh + n]; wk1 = w16[1 * Hh + n];
      wk2 = w16[2 * Hh + n]; wk3 = w16[3 * Hh + n];
    }
#pragma unroll
    for (int r = 0; r < 8; ++r) {
      int m = rb + rh + r;
      float e0 = acc[0][r] + bb0, e1 = acc[1][r] + bb1;
      float e2 = acc[2][r] + bb2, e3 = acc[3][r] + bb3;
      if (K16) { e0 += x16[r] * wk0; e1 += x16[r] * wk1; e2 += x16[r] * wk2; e3 += x16[r] * wk3; }
      float gi = sigf(e0), gf = sigf(e1), gg = tanha(e2), go = sigf(e3);
      float c = Cst[m * Hh + n];
      float cn = gf * c + gi * gg;
      Cst[m * Hh + n] = cn;
      Out[m * SO + n] = go * tanha(cn);
    }
  }
}

// Projection: HP(64x32) = S(64x128) @ Whr^T ; Whr is (32,128) row-major (ld=128, even).
__device__ __forceinline__ void proj_step(const float* __restrict__ S,
                                          const float* __restrict__ Whr,
                                          float* __restrict__ HP /*stride 36*/,
                                          float* __restrict__ OutG, int ldo) {
  const int lane = threadIdx.x & 31;
  const int wave = threadIdx.x >> 5;
  const int rb = wave * 16;
  const int m16 = lane & 15;
  const int rh = (lane >> 4) << 3;

  v8f acc0, acc1;
#pragma unroll
  for (int e = 0; e < 8; ++e) { acc0[e] = 0.0f; acc1[e] = 0.0f; }

#pragma unroll
  for (int kt = 0; kt < 32; ++kt) {
    v2f a = frag_a(S, 132, rb, 4 * kt, lane);
    v2f b0 = frag_b_gl(Whr, Hh, 0, 4 * kt, lane);
    v2f b1 = frag_b_gl(Whr, Hh, 16, 4 * kt, lane);
    acc0 = __builtin_amdgcn_wmma_f32_16x16x4_f32(false, a, false, b0, (short)0, acc0, false, false);
    acc1 = __builtin_amdgcn_wmma_f32_16x16x4_f32(false, a, false, b1, (short)0, acc1, false, false);
  }
#pragma unroll
  for (int r = 0; r < 8; ++r) {
    int m = rb + rh + r;
    float y0 = acc0[r], y1 = acc1[r];
    HP[m * 36 + m16] = y0;
    HP[m * 36 + 16 + m16] = y1;
    if (OutG) {
      OutG[(size_t)m * (size_t)ldo + m16] = y0;
      OutG[(size_t)m * (size_t)ldo + 16 + m16] = y1;
    }
  }
}

__global__ __launch_bounds__(NTHREADS, 1) void qlstm_fused_kernel(
    const float* __restrict__ meas0, const float* __restrict__ basr,
    const float* __restrict__ basi, const float* __restrict__ rho,
    const float* __restrict__ h0, const float* __restrict__ c0,
    const float* __restrict__ Wih_cell, const float* __restrict__ Whh_cell,
    const float* __restrict__ bih_cell, const float* __restrict__ bhh_cell,
    const float* __restrict__ Wp, const float* __restrict__ bp,
    const float* __restrict__ Wih0, const float* __restrict__ Whh0,
    const float* __restrict__ bih0, const float* __restrict__ bhh0,
    const float* __restrict__ Whr0, const float* __restrict__ Wih1,
    const float* __restrict__ Whh1, const float* __restrict__ bih1,
    const float* __restrict__ bhh1, const float* __restrict__ Whr1,
    float* __restrict__ out) {
  // ---- LDS (~272 KB < 320 KB WGP LDS) ----
  __shared__ float ls_x[TILE * 20];     // x_t (cols 0..16 used)
  __shared__ float ls_h[TILE * 132];    // cell hidden state
  __shared__ float ls_s[TILE * 132];    // gate output scratch (h_new / s)
  __shared__ float ls_hp0[TILE * 36];   // layer0 projected h
  __shared__ float ls_hp1[TILE * 36];   // layer1 projected h
  __shared__ float ls_cc[TILE * 128];   // cell c
  __shared__ float ls_c0[TILE * 128];   // layer0 c
  __shared__ float ls_c1[TILE * 128];   // layer1 c
  __shared__ float ls_v[TILE * 8];      // projector outputs
  __shared__ float ls_wt0[8 * 520 * 2]; // Wih_cell^T pair-swizzled (K=0..15)
  __shared__ float ls_wt1[8 * 520 * 2]; // Wih0^T pair-swizzled (K=0..15)
  __shared__ float ls_wp[8 * Hh];       // Wp
  __shared__ float ls_b0[512];          // bih_cell + bhh_cell
  __shared__ float ls_b1[512];          // bih0 + bhh0
  __shared__ float ls_b2[512];          // bih1 + bhh1
  __shared__ float ls_k16a[512];        // Wih_cell[:,16]
  __shared__ float ls_k16b[512];        // Wih0[:,16]

  const int tid = threadIdx.x;
  const size_t gb = (size_t)blockIdx.x * TILE;

  // ---- stage weights / biases into LDS ----
  for (int idx = tid; idx < 8 * 512; idx += NTHREADS) {
    int k2 = idx >> 9, n = idx & 511;
    ls_wt0[(k2 * 520 + n) * 2 + 0] = Wih_cell[n * DIN + 2 * k2 + 0];
    ls_wt0[(k2 * 520 + n) * 2 + 1] = Wih_cell[n * DIN + 2 * k2 + 1];
    ls_wt1[(k2 * 520 + n) * 2 + 0] = Wih0[n * DIN + 2 * k2 + 0];
    ls_wt1[(k2 * 520 + n) * 2 + 1] = Wih0[n * DIN + 2 * k2 + 1];
  }
  for (int idx = tid; idx < 512; idx += NTHREADS) {
    ls_k16a[idx] = Wih_cell[idx * DIN + 16];
    ls_k16b[idx] = Wih0[idx * DIN + 16];
    ls_b0[idx] = bih_cell[idx] + bhh_cell[idx];
    ls_b1[idx] = bih0[idx] + bhh0[idx];
    ls_b2[idx] = bih1[idx] + bhh1[idx];
  }
  for (int idx = tid; idx < 8 * Hh; idx += NTHREADS) ls_wp[idx] = Wp[idx];

  // ---- init recurrent state ----
  for (int idx = tid; idx < TILE * Hh; idx += NTHREADS) {
    int r = idx >> 7, k = idx & 127;
    ls_h[r * 132 + k] = h0[(gb + r) * Hh + k];
    ls_cc[idx] = c0[(gb + r) * Hh + k];
    ls_c0[idx] = 0.0f;
    ls_c1[idx] = 0.0f;
  }
  for (int idx = tid; idx < TILE * 36; idx += NTHREADS) {
    ls_hp0[idx] = 0.0f;
    ls_hp1[idx] = 0.0f;
  }
  if (tid < TILE) {  // build x0
    int r = tid;
    size_t rg = gb + r;
    ls_x[r * 20 + 0] = meas0[rg];
    const float* br = basr + rg * 8;
    const float* bi = basi + rg * 8;
#pragma unroll
    for (int q = 0; q < 2; ++q)
#pragma unroll
      for (int a = 0; a < 2; ++a)
#pragma unroll
        for (int b = 0; b < 2; ++b) {
          ls_x[r * 20 + 1 + q * 8 + a * 4 + b * 2 + 0] = br[q * 4 + a * 2 + b];
          ls_x[r * 20 + 1 + q * 8 + a * 4 + b * 2 + 1] = bi[q * 4 + a * 2 + b];
        }
    ls_x[r * 20 + 17] = 0.0f; ls_x[r * 20 + 18] = 0.0f; ls_x[r * 20 + 19] = 0.0f;
  }
  __syncthreads();

  for (int t = 0; t < Tt; ++t) {
    // ---- layer0: x(17) + hp0(32) -> s ; proj -> hp0 ----
    lstm_step<4, 20, 8, 36, 132, true, true>(ls_x, ls_hp0, ls_wt1, 0, Whh0, Pp,
                                             ls_b1, ls_k16b, ls_c0, ls_s);
    __syncthreads();
    proj_step(ls_s, Whr0, ls_hp0, nullptr, 0);
    __syncthreads();

    // ---- layer1: hp0(32) + hp1(32) -> s ; proj -> hp1 + global output ----
    lstm_step<8, 36, 8, 36, 132, false, false>(ls_hp0, ls_hp1, Wih1, Pp, Whh1, Pp,
                                               ls_b2, nullptr, ls_c1, ls_s);
    __syncthreads();
    proj_step(ls_s, Whr1, ls_hp1,
              out + gb * (size_t)(Tt * Pp) + (size_t)t * Pp, Tt * Pp);
    __syncthreads();

    if (t < Tt - 1) {
      // ---- measurement cell: x(17) + h(128) -> h_new (into ls_s; no in-place hazard) ----
      lstm_step<4, 20, 32, 132, 132, true, true>(ls_x, ls_h, ls_wt0, 0, Whh_cell, Hh,
                                                 ls_b0, ls_k16a, ls_cc, ls_s);
      __syncthreads();

      // projector v = h_new @ Wp^T + bp (2 threads/row, 4 outputs each) + copy s->h
      {
        int r = tid >> 1;
        int ob = (tid & 1) * 4;
        float a0 = bp[ob + 0], a1 = bp[ob + 1], a2 = bp[ob + 2], a3 = bp[ob + 3];
        for (int k = 0; k < Hh; ++k) {
          float hv = ls_s[r * 132 + k];
          a0 += hv * ls_wp[(ob + 0) * Hh + k];
          a1 += hv * ls_wp[(ob + 1) * Hh + k];
          a2 += hv * ls_wp[(ob + 2) * Hh + k];
          a3 += hv * ls_wp[(ob + 3) * Hh + k];
        }
        ls_v[r * 8 + ob + 0] = a0; ls_v[r * 8 + ob + 1] = a1;
        ls_v[r * 8 + ob + 2] = a2; ls_v[r * 8 + ob + 3] = a3;
        // wave-local copy of new hidden state
        int rb = (tid >> 5) * 16;
        int lane = tid & 31;
        for (int rr = 0; rr < 16; ++rr)
          for (int k = lane; k < Hh; k += 32)
            ls_h[(rb + rr) * 132 + k] = ls_s[(rb + rr) * 132 + k];
      }
      __syncthreads();

      // rank-1 measurement operators + trace with rho -> next x (1 thread/row)
      if (tid < TILE) {
        int r = tid;
        float Mre[2][2][2], Mim[2][2][2];
#pragma unroll
        for (int q = 0; q < 2; ++q) {
          float re[2] = {ls_v[r * 8 + q * 4 + 0], ls_v[r * 8 + q * 4 + 2]};
          float im[2] = {ls_v[r * 8 + q * 4 + 1], ls_v[r * 8 + q * 4 + 3]};
          float trc = re[0] * re[0] + im[0] * im[0] + re[1] * re[1] + im[1] * im[1];
          float inv = 1.0f / trc;
#pragma unroll
          for (int a = 0; a < 2; ++a)
#pragma unroll
            for (int b = 0; b < 2; ++b) {
              Mre[q][a][b] = (re[a] * re[b] + im[a] * im[b]) * inv;
              Mim[q][a][b] = (im[a] * re[b] - re[a] * im[b]) * inv;
            }
        }
        const float* rr = rho + (gb + r) * 32;  // [0][4][4] real, [1][4][4] imag
        float meas = 0.0f;
#pragma unroll
        for (int a = 0; a < 2; ++a)
#pragma unroll
          for (int cc2 = 0; cc2 < 2; ++cc2)
#pragma unroll
            for (int b = 0; b < 2; ++b)
#pragma unroll
              for (int d = 0; d < 2; ++d) {
                float pr = Mre[0][a][cc2] * Mre[1][b][d] - Mim[0][a][cc2] * Mim[1][b][d];
                float pi = Mre[0][a][cc2] * Mim[1][b][d] + Mim[0][a][cc2] * Mre[1][b][d];
                int ri = (cc2 * 2 + d) * 4 + a * 2 + b;
                meas += pr * rr[ri] - pi * rr[16 + ri];
              }
        ls_x[r * 20 + 0] = meas;
#pragma unroll
        for (int q = 0; q < 2; ++q)
#pragma unroll
          for (int a = 0; a < 2; ++a)
#pragma unroll
            for (int b = 0; b < 2; ++b) {
              ls_x[r * 20 + 1 + q * 8 + a * 4 + b * 2 + 0] = Mre[q][a][b];
              ls_x[r * 20 + 1 + q * 8 + a * 4 + b * 2 + 1] = Mim[q][a][b];
            }
      }
      __syncthreads();
    }
  }
}

extern "C" void kernel_launch(void* const* d_in, const int* in_sizes, int n_in,
                              void* d_out, int out_size, void* d_ws, size_t ws_size,
                              hipStream_t stream) {
  (void)in_sizes; (void)n_in; (void)d_ws; (void)ws_size; (void)out_size;
  const float* meas0    = (const float*)d_in[0];
  const float* basr     = (const float*)d_in[1];
  const float* basi     = (const float*)d_in[2];
  const float* rho      = (const float*)d_in[3];
  const float* h0       = (const float*)d_in[4];
  const float* c0       = (const float*)d_in[5];
  const float* Wih_cell = (const float*)d_in[6];
  const float* Whh_cell = (const float*)d_in[7];
  const float* bih_cell = (const float*)d_in[8];
  const float* bhh_cell = (const float*)d_in[9];
  const float* Wp       = (const float*)d_in[10];
  const float* bp       = (const float*)d_in[11];
  const float* Wih0     = (const float*)d_in[12];
  const float* Whh0     = (const float*)d_in[13];
  const float* bih0     = (const float*)d_in[14];
  const float* bhh0     = (const float*)d_in[15];
  const float* Whr0     = (const float*)d_in[16];
  const float* Wih1     = (const float*)d_in[17];
  const float* Whh1     = (const float*)d_in[18];
  const float* bih1     = (const float*)d_in[19];
  const float* bhh1     = (const float*)d_in[20];
  const float* Whr1     = (const float*)d_in[21];

  dim3 grid(B_TOTAL / TILE);
  dim3 block(NTHREADS);
  qlstm_fused_kernel<<<grid, block, 0, stream>>>(
      meas0, basr, basi, rho, h0, c0, Wih_cell, Whh_cell, bih_cell, bhh_cell, Wp, bp,
      Wih0, Whh0, bih0, bhh0, Whr0, Wih1, Whh1, bih1, bhh1, Whr1, (float*)d_out);
}